// Controller_18528488914997
// MI455X (gfx1250) — compile-verified
//
#include <hip/hip_runtime.h>
#include <hip/hip_bf16.h>
#include <math.h>

// CDNA5 / gfx1250, wave32. One workgroup of 8 waves.
typedef __attribute__((ext_vector_type(2))) float v2f;
typedef __attribute__((ext_vector_type(8))) float v8f;

#define HDIM 32
#define T_STEPS 20
#define NK_END 14      // steps 0..13 use kernel heads (8-way)
#define ND_END 17      // steps 14..16 use down heads (3-way), 17..19 up heads
#define NEGV (-1.0e9f)

__global__ __launch_bounds__(256) void controller_lstm_kernel(
    const float* __restrict__ w_ih,      // [128,32]
    const float* __restrict__ w_hh,      // [128,32]
    const float* __restrict__ b_ih,      // [128]
    const float* __restrict__ b_hh,      // [128]
    const float* __restrict__ g_emb,     // [1,32]
    const float* __restrict__ emb_kernel,// [8,32]
    const float* __restrict__ emb_down,  // [3,32]
    const float* __restrict__ emb_up,    // [3,32]
    const float* __restrict__ w_kernel,  // [8,32]
    const float* __restrict__ w_down,    // [3,32]
    const float* __restrict__ w_up,      // [3,32]
    const float* __restrict__ noise,     // [20,8]
    float* __restrict__ out)             // [2 + 20]
{
  // v = [x(32) ; h(32)], gates = [i f g o] (128)
  __shared__ __align__(16) float sV[64];
  __shared__ __align__(16) float sG[128];

  const int tid  = threadIdx.x;
  const int wave = tid >> 5;        // 0..7  -> 16-row block of gates
  const int lane = tid & 31;
  const int half = lane >> 4;       // 0: K=0,1 side, 1: K=2,3 side
  const int m16  = lane & 15;
  const int rowbase = wave * 16;
  const int row  = rowbase + m16;

  // ---- Preload resident A tiles: Wcat = [w_ih | w_hh], 128 rows x 64 cols ----
  // A 16x4 f32 layout: lanes 0-15 hold {K=0,K=1}, lanes 16-31 hold {K=2,K=3}.
  v2f a[16];
#pragma unroll
  for (int kc = 0; kc < 16; ++kc) {
    const int col0 = 4 * kc + 2 * half;     // even; pair never straddles ih/hh
    const float* Wsrc = (col0 < HDIM) ? (w_ih + row * HDIM + col0)
                                      : (w_hh + row * HDIM + (col0 - HDIM));
    v2f t; t[0] = Wsrc[0]; t[1] = Wsrc[1];
    a[kc] = t;
  }

  // ---- Bias folded into accumulator init (C layout: VGPR r = row rowbase+r+8*half)
  v8f cinit;
#pragma unroll
  for (int r = 0; r < 8; ++r) {
    const int rr = rowbase + r + 8 * half;
    cinit[r] = b_ih[rr] + b_hh[rr];
  }

  // ---- Initial state: x = g_emb, h = 0 ----
  if (tid < 64) sV[tid] = (tid < HDIM) ? g_emb[tid] : 0.0f;
  __syncthreads();

  float c_st = 0.0f;                 // cell state (meaningful in wave 0)
  float sum_lp = 0.0f, sum_ent = 0.0f;

  for (int t = 0; t < T_STEPS; ++t) {
    // ================= Stage A: gates = Wcat @ [x;h] + b  (all 8 waves) =====
    v8f acc = cinit;
#pragma unroll
    for (int kc = 0; kc < 16; ++kc) {
      // B 4x16: broadcast v chunk across all 16 columns (ds_load_b64 per lane)
      v2f bv = *(const v2f*)&sV[4 * kc + 2 * half];
      acc = __builtin_amdgcn_wmma_f32_16x16x4_f32(
          /*neg_a=*/false, a[kc], /*neg_b=*/false, bv,
          /*c_mod=*/(short)0, acc, /*reuse_a=*/false, /*reuse_b=*/false);
    }
    // D: VGPR r holds row rowbase+r (lanes 0-15) / rowbase+8+r (lanes 16-31),
    // identical across columns -> lanes 0 and 16 write the 16 rows.
    if (m16 == 0) {
      const int base = rowbase + 8 * half;
#pragma unroll
      for (int r = 0; r < 8; ++r) sG[base + r] = acc[r];
    }
    __syncthreads();

    // ================= Stage B: cell + heads + sampling (wave 0 only) =======
    if (wave == 0) {
      const float gi = sG[lane];
      const float gf = sG[HDIM + lane];
      const float gg = sG[2 * HDIM + lane];
      const float go = sG[3 * HDIM + lane];
      const float si = 1.0f / (1.0f + expf(-gi));
      const float sf = 1.0f / (1.0f + expf(-gf));
      const float so = 1.0f / (1.0f + expf(-go));
      c_st = sf * c_st + si * tanhf(gg);
      const float h = so * tanhf(c_st);   // lane j holds h[j]

      const int nv = (t < NK_END) ? 8 : 3;
      const float* P = (t < NK_END) ? w_kernel  : ((t < ND_END) ? w_down  : w_up);
      const float* E = (t < NK_END) ? emb_kernel : ((t < ND_END) ? emb_down : emb_up);

      // logits: lanes 0..nv-1 each do a 32-length dot, h shared via shuffles
      float dacc = 0.0f;
#pragma unroll
      for (int j = 0; j < HDIM; ++j) {
        const float hj = __shfl(h, j, 32);
        if (lane < nv) dacc += P[lane * HDIM + j] * hj;
      }
      const float logit = (lane < nv) ? dacc : NEGV;

      // broadcast the 8 logits to every lane, then redundant serial softmax
      float lg[8];
#pragma unroll
      for (int r = 0; r < 8; ++r) lg[r] = __shfl(logit, r, 32);

      float mx = NEGV;
#pragma unroll
      for (int r = 0; r < 8; ++r) if (r < nv) mx = fmaxf(mx, lg[r]);
      float se = 0.0f;
#pragma unroll
      for (int r = 0; r < 8; ++r) if (r < nv) se += expf(lg[r] - mx);
      const float lse = logf(se);

      // Gumbel-max categorical sample (masked lanes can never win)
      int   s    = 0;
      float best = -3.0e38f;
      float bl   = 0.0f;
#pragma unroll
      for (int r = 0; r < 8; ++r) {
        if (r < nv) {
          const float u   = noise[t * 8 + r];
          const float gum = -logf(-logf(u * (1.0f - 1e-6f) + 1e-7f));
          const float val = lg[r] + gum;
          if (val > best) { best = val; s = r; bl = lg[r]; }
        }
      }
      float ent = 0.0f;
#pragma unroll
      for (int r = 0; r < 8; ++r) {
        if (r < nv) {
          const float lp = lg[r] - mx - lse;
          ent -= expf(lp) * lp;
        }
      }
      sum_lp  += bl - mx - lse;
      sum_ent += ent;
      if (lane == 0) out[2 + t] = (float)s;

      // next-step input: x = emb[t][s], h -> LDS
      sV[lane]        = E[s * HDIM + lane];
      sV[HDIM + lane] = h;
    }
    __syncthreads();
  }

  if (tid == 0) { out[0] = sum_lp; out[1] = sum_ent; }
}

extern "C" void kernel_launch(void* const* d_in, const int* in_sizes, int n_in,
                              void* d_out, int out_size, void* d_ws, size_t ws_size,
                              hipStream_t stream) {
  (void)in_sizes; (void)n_in; (void)out_size; (void)d_ws; (void)ws_size;
  controller_lstm_kernel<<<1, 256, 0, stream>>>(
      (const float*)d_in[0],  (const float*)d_in[1],  (const float*)d_in[2],
      (const float*)d_in[3],  (const float*)d_in[4],  (const float*)d_in[5],
      (const float*)d_in[6],  (const float*)d_in[7],  (const float*)d_in[8],
      (const float*)d_in[9],  (const float*)d_in[10], (const float*)d_in[11],
      (float*)d_out);
}